// BTNetAmerican_79242146611470
// MI455X (gfx1250) — compile-verified
//
#include <hip/hip_runtime.h>
#include <cmath>

// MI455X / gfx1250, wave32. Binomial-tree American put, B=8192 strikes, N=256 steps.
// Compute-bound (~1e9 lane-ops, 64KB HBM traffic) -> keep everything in LDS/regs,
// use f32 WMMA (V_WMMA_F32_16X16X4_F32) for the bidiagonal combine, VALU for
// payoff/max (co-executes with the XDL pipe).

typedef float v2f __attribute__((ext_vector_type(2)));
typedef float v8f __attribute__((ext_vector_type(8)));

#define NSTEP 256
#define NODES 288   // 257 live nodes, padded to cover K-block over-reads (max idx 259)
#define BT    16    // strikes per wave

__global__ __launch_bounds__(32)
void btree_american_wmma(const float* __restrict__ kin, float* __restrict__ out,
                         float w0, float w1, float uscale, float alpha)
{
    // V[node][strike]  (node-major, 16 strikes wide): 288*16*4 = 18KB
    __shared__ float V[NODES * BT];
    __shared__ float PB[NODES];          // S0*exp(alpha*(2j-(N-1))) payoff base

    const int lane = threadIdx.x;        // 0..31 (one wave per workgroup)
    const int l15  = lane & 15;          // strike within tile / matrix row
    const int hi   = lane >> 4;          // lane half selects K/M sub-block
    const int b0   = blockIdx.x * BT;

    const float kv = kin[b0 + l15];      // this lane's strike

    // Terminal layer v0[j] = relu(k - S0*exp(alpha*(2j - N))); finite junk in pad.
    for (int j = hi; j < NODES; j += 2) {
        float st = 100.0f * expf(alpha * (float)(2 * j - NSTEP));
        V[j * BT + l15] = fmaxf(kv - st, 0.0f);
    }
    for (int j = lane; j < NODES; j += 32) {
        PB[j] = 100.0f * expf(alpha * (float)(2 * j - (NSTEP - 1)));
    }
    __syncthreads();   // single-wave WG: barrier -> S_NOP, keeps ds ordering fences

    // A = W^T band matrix in WMMA A-layout (16x4 f32: lanes 0-15 = rows M,
    // VGPR0 holds K = 0|2 by lane half, VGPR1 holds K = 1|3).
    // W^T[m,k] = w0 if k==m, w1 if k==m+1 (k global-in-tile = 4t+kk).
    v2f Av[5];
#pragma unroll
    for (int t = 0; t < 5; ++t) {
        int k0 = 4 * t + 2 * hi;
        int k1 = k0 + 1;
        float e0 = (k0 == l15) ? w0 : ((k0 == l15 + 1) ? w1 : 0.0f);
        float e1 = (k1 == l15) ? w0 : ((k1 == l15 + 1) ? w1 : 0.0f);
        Av[t][0] = e0;
        Av[t][1] = e1;
    }

    float scale = 1.0f;   // exp(alpha * i)
    for (int i = 0; i < NSTEP; ++i) {
        const int n_out  = NSTEP - i;            // shrinking valid region
        const int ntiles = (n_out + 15) >> 4;

        for (int g = 0; g < ntiles; ++g) {
            const int base = g << 4;

            // B = V tile in WMMA B-layout (4x16 f32): per lane, elem0 = node
            // base+4t+2*hi, elem1 = next node; strike = lane&15. LDS per-lane
            // addressing produces the layout directly -- no permutes.
            v2f Bv[5];
#pragma unroll
            for (int t = 0; t < 5; ++t) {
                int kb = base + 4 * t + 2 * hi;
                Bv[t][0] = V[kb * BT + l15];
                Bv[t][1] = V[(kb + 1) * BT + l15];
            }

            // cont(16 out-nodes x 16 strikes) = sum_t A_t x B_t, K=4 chain (17 live K)
            v8f c = {};
#pragma unroll
            for (int t = 0; t < 5; ++t) {
                c = __builtin_amdgcn_wmma_f32_16x16x4_f32(
                        /*neg_a=*/false, Av[t],
                        /*neg_b=*/false, Bv[t],
                        /*c_mod=*/(short)0, c,
                        /*reuse_a=*/false, /*reuse_b=*/false);
            }

            // American exercise: max(cont, k - scale*PB[node]); write back in place.
            // D layout: VGPR r -> node base+r (lanes 0-15) / base+r+8 (lanes 16-31).
#pragma unroll
            for (int r = 0; r < 8; ++r) {
                int node  = base + r + 8 * hi;
                float pay = fmaf(-scale, PB[node], kv);
                V[node * BT + l15] = fmaxf(c[r], pay);
            }
        }
        scale *= uscale;
        __syncthreads();
    }

    if (hi == 0) out[b0 + l15] = V[l15];   // v[:,0]
}

extern "C" void kernel_launch(void* const* d_in, const int* in_sizes, int n_in,
                              void* d_out, int out_size, void* d_ws, size_t ws_size,
                              hipStream_t stream) {
    (void)n_in; (void)d_ws; (void)ws_size; (void)out_size;
    const float* kin = (const float*)d_in[0];
    float* out = (float*)d_out;
    const int B = in_sizes[0];            // 8192 strikes

    const double S0g = 100.0, SIG = 0.2, Tg = 1.0, T0g = 0.0, Rg = 0.05;
    (void)S0g;
    const double dt   = (Tg - T0g) / (double)NSTEP;
    const double sq   = sqrt(dt);
    const double u    = exp(SIG * sq);
    const double d    = exp(-SIG * sq);
    const double disc = exp(-Rg * dt);
    const float w0    = (float)((disc * u - 1.0) / (u - d));
    const float w1    = (float)((1.0 - disc * d) / (u - d));
    const float us    = (float)u;
    const float alpha = (float)(SIG * sq);

    const int nblocks = B / BT;           // 512 independent single-wave workgroups
    btree_american_wmma<<<nblocks, 32, 0, stream>>>(kin, out, w0, w1, us, alpha);
}